// RRSVM_v2_M_51840255262889
// MI455X (gfx1250) — compile-verified
//
#include <hip/hip_runtime.h>
#include <cstdint>

// RRSVM rank-weighted pooling, MI455X (gfx1250).
// x: [32,64,112,112] f32, s: [64,3,3] f32
// d_out = out [32,64,56,56] f32  ++  order [32,64,56,56,9] int64.
//
// Store-dominated memory-bound kernel: ~591 MB HBM traffic (103 MB x read,
// 26 MB out, 462 MB int64 order) -> ~25us floor @ 23.3 TB/s. No GEMM
// structure exists (per-window sort), so the CDNA5-specific path is the
// async global->LDS copy engine (global_load_async_to_lds_b32 / ASYNCcnt)
// for coalesced input staging, plus LDS-staged fully coalesced b128
// streaming of the dominant int64 `order` output.

#define B_  32
#define C_  64
#define H_  112
#define W_  112
#define HO_ 56
#define WO_ 56
#define K_  9
#define ROWP 120  // LDS row pitch in floats; cols 0..112 used (input col-1)

__global__ __launch_bounds__(64)
void rrsvm_kernel(const float* __restrict__ x,
                  const float* __restrict__ s,
                  float* __restrict__ out,
                  long long* __restrict__ order)
{
    __shared__ float sm[3 * ROWP];                                    // 3 input rows + halo
    __shared__ __attribute__((aligned(16))) long long sord[WO_ * K_]; // 504 int64 staging

    const int tid = (int)threadIdx.x;
    const int blk = (int)blockIdx.x;
    const int ho  = blk % HO_;
    const int bc  = blk / HO_;          // b*C + c
    const int c   = bc % C_;
    const int h0  = ho * 2 - 1;         // top input row of the 3-row band (may be -1)

    // ---- 1) stage 3 input rows: async global->LDS; uniform per-row branch ----
    const uint64_t xbase = (uint64_t)(uintptr_t)(x + (size_t)bc * (H_ * W_));
    #pragma unroll
    for (int r = 0; r < 3; ++r) {
        const int h = h0 + r;
        if (h >= 0 && h < H_) {         // uniform (scalar) condition
            const uint32_t row_goff = (uint32_t)(h * W_) * 4u;
            for (int w = tid; w < W_; w += 64) {
                const uint32_t lds_b = (uint32_t)(uintptr_t)&sm[r * ROWP + (w + 1)];
                const uint32_t goff  = row_goff + (uint32_t)w * 4u;
                asm volatile("global_load_async_to_lds_b32 %0, %1, %2"
                             :: "v"(lds_b), "v"(goff), "s"(xbase)
                             : "memory");
            }
        } else {                        // only ho==0, r==0: zero the pad row
            for (int i = tid; i < 113; i += 64)
                sm[r * ROWP + i] = 0.0f;
        }
    }
    if (tid < 3)                        // left pad column (input col -1), all 3 rows
        sm[tid * ROWP] = 0.0f;
    asm volatile("s_wait_asynccnt 0" ::: "memory");
    __syncthreads();

    // ---- 2) per-pixel: stable descending sort of 9 taps + rank-weighted sum ----
    if (tid < WO_) {
        const int wo = tid;
        float v[K_];
        int   id[K_];
        #pragma unroll
        for (int kh = 0; kh < 3; ++kh) {
            #pragma unroll
            for (int kw = 0; kw < 3; ++kw) {
                v[kh * 3 + kw]  = sm[kh * ROWP + (2 * wo + kw)];
                id[kh * 3 + kw] = kh * 3 + kw;
            }
        }
        // Branchless insertion sort: strict '>' swap => stable descending,
        // exactly matching stable argsort(-patches) with index tiebreak.
        #pragma unroll
        for (int i = 1; i < K_; ++i) {
            #pragma unroll
            for (int j = i; j > 0; --j) {
                const bool  sw = v[j] > v[j - 1];
                const float va = sw ? v[j]     : v[j - 1];
                const float vb = sw ? v[j - 1] : v[j];
                const int   ia = sw ? id[j]     : id[j - 1];
                const int   ib = sw ? id[j - 1] : id[j];
                v[j - 1]  = va;  v[j]  = vb;
                id[j - 1] = ia;  id[j] = ib;
            }
        }
        float acc = 0.0f;
        #pragma unroll
        for (int k = 0; k < K_; ++k)
            acc += v[k] * s[c * K_ + k];               // uniform -> scalar loads
        out[((size_t)bc * HO_ + ho) * WO_ + wo] = acc; // coalesced b32

        #pragma unroll
        for (int k = 0; k < K_; ++k)
            sord[wo * K_ + k] = (long long)id[k];
    }
    __syncthreads();

    // ---- 3) stream int64 order indices: 252 coalesced 16B stores per row ----
    long long* obase = order + ((size_t)bc * HO_ + ho) * (WO_ * K_); // 4032B-aligned
    const int4* srcv = (const int4*)sord;
    int4*       dstv = (int4*)obase;
    #pragma unroll
    for (int i = tid; i < (WO_ * K_) / 2; i += 64)
        dstv[i] = srcv[i];
}

extern "C" void kernel_launch(void* const* d_in, const int* in_sizes, int n_in,
                              void* d_out, int out_size, void* d_ws, size_t ws_size,
                              hipStream_t stream) {
    const float* x = (const float*)d_in[0];   // [32,64,112,112]
    const float* s = (const float*)d_in[1];   // [64,3,3]
    float* out = (float*)d_out;
    const size_t nout = (size_t)B_ * C_ * HO_ * WO_;
    long long* order = (long long*)(out + nout);  // byte offset nout*4, 16B-aligned

    dim3 grid((unsigned)(B_ * C_ * HO_));     // 114,688 blocks: (b,c,ho)
    rrsvm_kernel<<<grid, 64, 0, stream>>>(x, s, out, order);
}